// Net_19000935317562
// MI455X (gfx1250) — compile-verified
//
#include <hip/hip_runtime.h>
#include <math.h>

typedef __attribute__((ext_vector_type(2))) float v2f;
typedef __attribute__((ext_vector_type(4))) float v4f;
typedef __attribute__((ext_vector_type(8))) float v8f;

#define NPIX 256
#define NVIS 50000

// cell size in radians: 0.005 arcsec
#define CELL_RAD  2.42406840554768e-8f
#define TWO_PI    6.28318530717958647692f

// ---------------- Kernel 1: softplus ----------------
__global__ __launch_bounds__(256) void softplus_kernel(const float* __restrict__ in,
                                                       float* __restrict__ out) {
    int i = blockIdx.x * blockDim.x + threadIdx.x;
    float x = in[i];
    out[i] = (x > 20.0f) ? x : log1pf(expf(x));
}

// ------- Kernel 2: 3x3 Hann conv (zero pad) + pack into WMMA-B layout -------
// Packed B operand: entry(ytile, chunk, lane) = float2{ img[y][x0], img[y][x0+1] }
// with y = ytile*16 + (lane&15), x0 = chunk*4 + ((lane>>4)<<1).
// Pixel (y, x) maps to: ytile=y>>4, chunk=x>>2, lane=((x&2)<<3)|(y&15), slot=x&1.
__global__ __launch_bounds__(256) void hann_pack_kernel(const float* __restrict__ s,
                                                        float* __restrict__ Bpk) {
    int i = blockIdx.x * blockDim.x + threadIdx.x;
    int y = i >> 8;
    int x = i & 255;
    const float h0 = 0.25f, h1 = 0.5f;
    float acc = 0.0f;
#pragma unroll
    for (int dy = -1; dy <= 1; ++dy) {
        int yy = y + dy;
        if ((unsigned)yy >= (unsigned)NPIX) continue;
        float hy = (dy == 0) ? h1 : h0;
        float row = 0.0f;
#pragma unroll
        for (int dx = -1; dx <= 1; ++dx) {
            int xx = x + dx;
            if ((unsigned)xx >= (unsigned)NPIX) continue;
            float hx = (dx == 0) ? h1 : h0;
            row += hx * s[yy * NPIX + xx];
        }
        acc += hy * row;
    }
    int ytile = y >> 4;
    int chunk = x >> 2;
    int lane  = ((x & 2) << 3) | (y & 15);
    int slot  = x & 1;
    Bpk[(((ytile * 64 + chunk) * 32 + lane) << 1) | slot] = acc;
}

// ---------------- Kernel 3: NUDFT via f32 WMMA ----------------
// One workgroup (4 waves / 128 threads) per block of 16 visibilities.
// t[k,y] = sum_x a[k,x] * img[y,x]  via V_WMMA_F32_16X16X4_F32 (re & im GEMMs),
// then vis[k] = sum_y t[k,y] * b[k,y] * cell^2.
__global__ __launch_bounds__(128) void nudft_wmma_kernel(const v2f* __restrict__ Bpk,
                                                         const float* __restrict__ uu,
                                                         const float* __restrict__ vv,
                                                         float* __restrict__ out) {
    // A-operand tiles in exact WMMA per-lane layout:
    //   entry(chunk, lane): M = lane&15, x = chunk*4 + ((lane>>4)*2) (+0,+1)
    //   float4 = (re0, re1, im0, im1)
    __shared__ v4f   A_lds[64][32];          // 32 KB
    __shared__ float wu_lds[16];
    __shared__ float wv_lds[16];
    __shared__ float part[4][16][2];

    const int tid    = threadIdx.x;
    const int lane   = tid & 31;
    const int wave   = tid >> 5;
    const int kbase  = blockIdx.x * 16;

    if (tid < 16) {
        wu_lds[tid] = -TWO_PI * 1.0e3f * uu[kbase + tid];   // kilolambda -> lambda
        wv_lds[tid] = -TWO_PI * 1.0e3f * vv[kbase + tid];
    }
    __syncthreads();

    // Precompute A phase tiles cooperatively (shared by all 4 waves).
    for (int idx = tid; idx < 64 * 32; idx += 128) {
        int chunk = idx >> 5;
        int l     = idx & 31;
        int x0    = chunk * 4 + ((l >> 4) << 1);
        float w   = wu_lds[l & 15];
        float c0  = ((float)x0 - 128.0f) * CELL_RAD;
        float c1  = ((float)(x0 + 1) - 128.0f) * CELL_RAD;
        float s0, cc0, s1, cc1;
        __sincosf(w * c0, &s0, &cc0);
        __sincosf(w * c1, &s1, &cc1);
        A_lds[chunk][l] = (v4f){cc0, cc1, s0, s1};
    }
    __syncthreads();

    const int yl   = lane & 15;          // N index within a tile
    const int Moff = (lane >> 4) << 3;   // C/D rows this lane holds: Moff..Moff+7

    float vr[8], vi[8];
#pragma unroll
    for (int r = 0; r < 8; ++r) { vr[r] = 0.0f; vi[r] = 0.0f; }

    // Warm the packed-B stream (global_prefetch_b8)
    __builtin_prefetch(Bpk + (wave * 4) * (64 * 32) + lane, 0, 0);

    for (int yt = 0; yt < 4; ++yt) {
        const int ytile = wave * 4 + yt;
        const int yg    = ytile * 16 + yl;          // this lane's y column
        const v2f* __restrict__ Bp = Bpk + ytile * (64 * 32) + lane;

        v8f acc_re = {};
        v8f acc_im = {};
#pragma unroll 4
        for (int chunk = 0; chunk < 64; ++chunk) {
            v4f a4 = A_lds[chunk][lane];
            v2f a_re = {a4.x, a4.y};
            v2f a_im = {a4.z, a4.w};
            v2f b = Bp[chunk * 32];                 // single coalesced b64 load
            acc_re = __builtin_amdgcn_wmma_f32_16x16x4_f32(
                false, a_re, false, b, (short)0, acc_re, false, false);
            acc_im = __builtin_amdgcn_wmma_f32_16x16x4_f32(
                false, a_im, false, b, (short)0, acc_im, false, false);
        }

        // Stage 2: t[k,y] * exp(-2*pi*i * v_k * y')
        float cy = ((float)yg - 128.0f) * CELL_RAD;
#pragma unroll
        for (int r = 0; r < 8; ++r) {
            float sb, cb;
            __sincosf(wv_lds[Moff + r] * cy, &sb, &cb);
            float tr = acc_re[r], ti = acc_im[r];
            vr[r] += tr * cb - ti * sb;
            vi[r] += tr * sb + ti * cb;
        }
    }

    // Reduce across the 16 lanes of each half-wave (lanes sharing the same M set).
#pragma unroll
    for (int m = 1; m <= 8; m <<= 1) {
#pragma unroll
        for (int r = 0; r < 8; ++r) {
            vr[r] += __shfl_xor(vr[r], m, 32);
            vi[r] += __shfl_xor(vi[r], m, 32);
        }
    }
    if (yl == 0) {
#pragma unroll
        for (int r = 0; r < 8; ++r) {
            part[wave][Moff + r][0] = vr[r];
            part[wave][Moff + r][1] = vi[r];
        }
    }
    __syncthreads();

    // Cross-wave reduction + scaled store (complex64 interleaved).
    if (tid < 32) {
        int kk = tid >> 1, comp = tid & 1;
        float s = part[0][kk][comp] + part[1][kk][comp] +
                  part[2][kk][comp] + part[3][kk][comp];
        out[(kbase + kk) * 2 + comp] = s * (CELL_RAD * CELL_RAD);
    }
}

extern "C" void kernel_launch(void* const* d_in, const int* in_sizes, int n_in,
                              void* d_out, int out_size, void* d_ws, size_t ws_size,
                              hipStream_t stream) {
    const float* base_cube = (const float*)d_in[0];   // (1,256,256) f32
    const float* uu        = (const float*)d_in[1];   // (50000,) f32 [kilolambda]
    const float* vv        = (const float*)d_in[2];   // (50000,) f32 [kilolambda]
    float*       out       = (float*)d_out;           // 50000 complex64 interleaved

    float* s_buf = (float*)d_ws;                 // 256KB: softplus image
    float* Bpk   = s_buf + NPIX * NPIX;          // 256KB: conv result, WMMA-B packed

    softplus_kernel<<<NPIX * NPIX / 256, 256, 0, stream>>>(base_cube, s_buf);
    hann_pack_kernel<<<NPIX * NPIX / 256, 256, 0, stream>>>(s_buf, Bpk);
    nudft_wmma_kernel<<<NVIS / 16, 128, 0, stream>>>((const v2f*)Bpk, uu, vv, out);
}